// SpatialControlLoss_39333310497232
// MI455X (gfx1250) — compile-verified
//
#include <hip/hip_runtime.h>
#include <hip/hip_bf16.h>
#include <math.h>

// Shapes are fixed by the reference: B=32, C=16, H=256, W=256, n_obj=16.
// Workspace layout (floats): [0]=sum((p-t)^2), [1]=sum((sa-ts)^2), [2..33]=per-batch count.

typedef float v2f __attribute__((ext_vector_type(2)));
typedef float v8f __attribute__((ext_vector_type(8)));

// Exact 32-lane f32 sum using V_WMMA_F32_16X16X4_F32.
// A: VGPR0 = v (lanes 0-15 -> A[m][0], lanes 16-31 -> A[m][2]), VGPR1 = 0.
// B: all ones (layout-agnostic). D[m][n] = v_m + v_{m+16}.
// Lane L<16 holds D[0..7][L] in 8 VGPRs; lane L>=16 holds D[8..15][L-16].
// Sum of own 8 VGPRs = half-wave total; xor-16 shuffle adds the other half.
__device__ __forceinline__ float wave_sum32(float v) {
  v2f a; a[0] = v;    a[1] = 0.0f;
  v2f bm; bm[0] = 1.0f; bm[1] = 1.0f;
  v8f c = {};
  v8f d = __builtin_amdgcn_wmma_f32_16x16x4_f32(
      /*neg_a=*/false, a, /*neg_b=*/false, bm,
      /*c_mod=*/(short)0, c, /*reuse_a=*/false, /*reuse_b=*/false);
  float s = d[0] + d[1] + d[2] + d[3] + d[4] + d[5] + d[6] + d[7];
  s += __shfl_xor(s, 16, 32);
  return s;  // full-wave total in every lane
}

__global__ void scl_init(float* __restrict__ acc) {
  int i = threadIdx.x;
  if (i < 34) acc[i] = 0.0f;
}

__global__ __launch_bounds__(256) void scl_main(const float* __restrict__ pred,
                                                const float* __restrict__ targ,
                                                const float* __restrict__ layout,
                                                float* __restrict__ acc) {
  const int tid  = threadIdx.x;
  const int b    = blockIdx.y;                 // batch 0..31
  const int pixg = blockIdx.x * 256 + tid;     // float4 group id, 0..16383

  __shared__ int s_xs[16], s_xe[16], s_ys[16], s_ye[16];
  if (tid < 16) {
    const float* L = layout + ((size_t)b * 16 + (size_t)tid) * 4;
    float x = L[0], y = L[1], w = L[2], h = L[3];
    bool valid = (x > 0.0f) && (y > 0.0f);
    s_xs[tid] = valid ? (int)floorf(x * 256.0f) : 0x7FFFFFFF;  // kills mask if invalid
    s_xe[tid] = (int)floorf((x + w) * 256.0f);
    s_ys[tid] = (int)floorf(y * 256.0f);
    s_ye[tid] = (int)floorf((y + h) * 256.0f);
  }
  __syncthreads();

  // Streaming pass: 16 channels x (pred,targ) float4 loads, fully unrolled.
  const float4* p4 = (const float4*)pred;
  const float4* t4 = (const float4*)targ;
  const size_t base = (size_t)b * 16u * 16384u + (size_t)pixg;  // 16384 float4 per image

  float a0 = 0.f, a1 = 0.f, a2 = 0.f, a3 = 0.f;  // sum |pred| over channels, per pixel
  float bsum = 0.f;                               // sum (p-t)^2 over channels & 4 pixels
#pragma unroll
  for (int c = 0; c < 16; ++c) {
    float4 p = p4[base + (size_t)c * 16384u];
    float4 t = t4[base + (size_t)c * 16384u];
    float d0 = p.x - t.x, d1 = p.y - t.y, d2 = p.z - t.z, d3 = p.w - t.w;
    bsum = fmaf(d0, d0, bsum); bsum = fmaf(d1, d1, bsum);
    bsum = fmaf(d2, d2, bsum); bsum = fmaf(d3, d3, bsum);
    a0 += fabsf(p.x); a1 += fabsf(p.y); a2 += fabsf(p.z); a3 += fabsf(p.w);
  }

  // Box-union mask for this row / 4 columns.
  const int row  = pixg >> 6;          // pix0 >> 8, pix0 = pixg*4
  const int col0 = (pixg & 63) << 2;
  int in0 = 0, in1 = 0, in2 = 0, in3 = 0;
#pragma unroll
  for (int o = 0; o < 16; ++o) {
    int rok = (row >= s_ys[o]) & (row < s_ye[o]);
    int lo = s_xs[o], hi = s_xe[o];
    in0 |= rok & (col0 + 0 >= lo) & (col0 + 0 < hi);
    in1 |= rok & (col0 + 1 >= lo) & (col0 + 1 < hi);
    in2 |= rok & (col0 + 2 >= lo) & (col0 + 2 < hi);
    in3 |= rok & (col0 + 3 >= lo) & (col0 + 3 < hi);
  }

  const float inv16 = 0.0625f;
  float sa0 = a0 * inv16, sa1 = a1 * inv16, sa2 = a2 * inv16, sa3 = a3 * inv16;
  float e0 = sa0 - (float)in0, e1 = sa1 - (float)in1;
  float e2 = sa2 - (float)in2, e3 = sa3 - (float)in3;
  float ssum = e0 * e0 + e1 * e1 + e2 * e2 + e3 * e3;
  float cnt  = (sa0 > 0.5f ? 1.f : 0.f) + (sa1 > 0.5f ? 1.f : 0.f) +
               (sa2 > 0.5f ? 1.f : 0.f) + (sa3 > 0.5f ? 1.f : 0.f);

  // Wave-level reductions via WMMA (EXEC is all-ones here: no divergence).
  float bw = wave_sum32(bsum);
  float sw = wave_sum32(ssum);
  float cw = wave_sum32(cnt);

  if ((tid & 31) == 0) {
    atomicAdd(&acc[0], bw);
    atomicAdd(&acc[1], sw);
    atomicAdd(&acc[2 + b], cw);   // integer-valued f32 adds: exact (<= 65536 per batch)
  }
}

__global__ void scl_finalize(const float* __restrict__ acc,
                             const float* __restrict__ obj_counts,
                             float* __restrict__ out) {
  int lane = threadIdx.x;  // exactly 32 threads == 32 batches
  float d = acc[2 + lane] - obj_counts[lane];
  float csum = wave_sum32(d * d);
  if (lane == 0) {
    float base_loss     = acc[0] * (1.0f / 33554432.0f);  // / (32*16*256*256)
    float spatial_loss  = acc[1] * (1.0f / 2097152.0f);   // / (32*256*256)
    float counting_loss = csum * (1.0f / 32.0f);
    out[0] = base_loss + spatial_loss + 0.5f * counting_loss;
  }
}

extern "C" void kernel_launch(void* const* d_in, const int* in_sizes, int n_in,
                              void* d_out, int out_size, void* d_ws, size_t ws_size,
                              hipStream_t stream) {
  (void)in_sizes; (void)n_in; (void)out_size; (void)ws_size;
  const float* pred   = (const float*)d_in[0];
  const float* targ   = (const float*)d_in[1];
  const float* layout = (const float*)d_in[2];
  const float* objc   = (const float*)d_in[3];
  float* out = (float*)d_out;
  float* acc = (float*)d_ws;  // 34 floats

  scl_init<<<1, 64, 0, stream>>>(acc);
  scl_main<<<dim3(64, 32), 256, 0, stream>>>(pred, targ, layout, acc);
  scl_finalize<<<1, 32, 0, stream>>>(acc, objc, out);
}